// MdsOWLoss_cov_73770358276631
// MI455X (gfx1250) — compile-verified
//
#include <hip/hip_runtime.h>
#include <hip/hip_bf16.h>
#include <stdint.h>

// Problem constants (from reference setup_inputs)
#define BB   8
#define DD   128
#define HH   256
#define WW   256
#define KK   64
#define HWC  65536            // H*W
#define NQ   131072           // total float4-pixel quads = B*H*W/4
#define STRIDE 132            // padded LDS row stride in floats (132 mod 64 = 4)
#define NBLK 512              // must stay a multiple of 64 (finalize WMMA tiles)
#define NTHR 256
#define EPSV 1e-8f

static_assert(NBLK % 64 == 0, "finalize assumes partials are a multiple of 64");

typedef __attribute__((ext_vector_type(2))) float v2f;
typedef __attribute__((ext_vector_type(8))) float v8f;

// ---- CDNA5 async global->LDS copy (ASYNCcnt path) ----------------------------
__device__ __forceinline__ void async_copy_b128(uint32_t lds_byte_off,
                                                uint32_t glb_byte_off,
                                                const void* base) {
  // GVS mode: mem = SADDR(64b) + VADDR(32b); VDST holds LDS byte address.
  asm volatile("global_load_async_to_lds_b128 %0, %1, %2"
               :: "v"(lds_byte_off), "v"(glb_byte_off), "s"(base)
               : "memory");
}
__device__ __forceinline__ void wait_async0() {
  asm volatile("s_wait_asynccnt 0" ::: "memory");
}

__device__ __forceinline__ float wave_sum32(float v) {
  #pragma unroll
  for (int m = 16; m >= 1; m >>= 1) v += __shfl_xor(v, m, 32);
  return v;
}

// ---- Kernel 1: inv_ev[k*128+d] = 1/(var[k,d,d] + eps) ------------------------
__global__ void mds_inv_diag(const float* __restrict__ var,
                             float* __restrict__ inv_ev) {
  int i = blockIdx.x * blockDim.x + threadIdx.x;   // 0..8191
  int k = i >> 7, d = i & 127;
  inv_ev[i] = 1.0f / (var[(size_t)k * (DD * DD) + (size_t)d * (DD + 1)] + EPSV);
}

// ---- Kernel 2: fused argmax + masked L1/var streaming reduce ----------------
__global__ void __launch_bounds__(NTHR, 1)
mds_main(const float* __restrict__ emb,      // [B,D,H,W]
         const float* __restrict__ logits,   // [B,K,H,W]
         const float* __restrict__ pf,       // [K,D]
         const float* __restrict__ pc,       // [K]
         const float* __restrict__ inv_ev,   // [K,D] (ws)
         float* __restrict__ partS,          // [NBLK]
         float* __restrict__ partC) {        // [NBLK]
  extern __shared__ float smem[];
  float* mav_t  = smem;                       // KK*STRIDE
  float* inv_t  = smem + KK * STRIDE;         // KK*STRIDE
  float* validt = inv_t + KK * STRIDE;        // KK
  float* redS   = validt + KK;                // 8
  float* redC   = redS + 8;                   // 8

  const int tid = threadIdx.x;

  // Async-load both gather tables into padded LDS rows (32 b128 chunks / row).
  for (int c = tid; c < (KK * DD / 4); c += NTHR) {
    int g  = c >> 5;               // row (class)
    int dc = (c & 31) << 2;        // d offset within row
    uint32_t goff = (uint32_t)c * 16u;
    async_copy_b128((uint32_t)(uintptr_t)&mav_t[g * STRIDE + dc], goff, pf);
    async_copy_b128((uint32_t)(uintptr_t)&inv_t[g * STRIDE + dc], goff, inv_ev);
  }
  if (tid < KK) validt[tid] = (pc[tid] > 0.0f) ? 1.0f : 0.0f;
  wait_async0();
  __syncthreads();

  float tsum = 0.0f, tcnt = 0.0f;

  for (int q = blockIdx.x * NTHR + tid; q < NQ; q += gridDim.x * NTHR) {
    int b   = q >> 14;                 // quads per image = HW/4 = 16384
    int rem = (q & 16383) << 2;        // pixel offset inside image (w-quad)

    // ---- argmax over K=64 logits for 4 adjacent pixels (b128 loads) ----
    const float4* lg = (const float4*)(logits + (size_t)b * KK * HWC + rem);
    float4 best = lg[0];
    int g0 = 0, g1 = 0, g2 = 0, g3 = 0;
    for (int k = 1; k < KK; ++k) {
      float4 v = lg[(size_t)k * (HWC / 4)];
      if (v.x > best.x) { best.x = v.x; g0 = k; }
      if (v.y > best.y) { best.y = v.y; g1 = k; }
      if (v.z > best.z) { best.z = v.z; g2 = k; }
      if (v.w > best.w) { best.w = v.w; g3 = k; }
    }

    // ---- masked L1 / variance accumulation over D=128 ----
    const float4* xe = (const float4*)(emb + (size_t)b * DD * HWC + rem);
    float a0 = 0.0f, a1 = 0.0f, a2 = 0.0f, a3 = 0.0f;
    #pragma unroll 8
    for (int d = 0; d < DD; d += 4) {
      float4 x0 = xe[(size_t)(d + 0) * (HWC / 4)];
      float4 x1 = xe[(size_t)(d + 1) * (HWC / 4)];
      float4 x2 = xe[(size_t)(d + 2) * (HWC / 4)];
      float4 x3 = xe[(size_t)(d + 3) * (HWC / 4)];
      float4 m0 = *(const float4*)&mav_t[g0 * STRIDE + d];
      float4 i0 = *(const float4*)&inv_t[g0 * STRIDE + d];
      float4 m1 = *(const float4*)&mav_t[g1 * STRIDE + d];
      float4 i1 = *(const float4*)&inv_t[g1 * STRIDE + d];
      float4 m2 = *(const float4*)&mav_t[g2 * STRIDE + d];
      float4 i2 = *(const float4*)&inv_t[g2 * STRIDE + d];
      float4 m3 = *(const float4*)&mav_t[g3 * STRIDE + d];
      float4 i3 = *(const float4*)&inv_t[g3 * STRIDE + d];
      a0 = fmaf(fabsf(x0.x - m0.x), i0.x, a0);
      a0 = fmaf(fabsf(x1.x - m0.y), i0.y, a0);
      a0 = fmaf(fabsf(x2.x - m0.z), i0.z, a0);
      a0 = fmaf(fabsf(x3.x - m0.w), i0.w, a0);
      a1 = fmaf(fabsf(x0.y - m1.x), i1.x, a1);
      a1 = fmaf(fabsf(x1.y - m1.y), i1.y, a1);
      a1 = fmaf(fabsf(x2.y - m1.z), i1.z, a1);
      a1 = fmaf(fabsf(x3.y - m1.w), i1.w, a1);
      a2 = fmaf(fabsf(x0.z - m2.x), i2.x, a2);
      a2 = fmaf(fabsf(x1.z - m2.y), i2.y, a2);
      a2 = fmaf(fabsf(x2.z - m2.z), i2.z, a2);
      a2 = fmaf(fabsf(x3.z - m2.w), i2.w, a2);
      a3 = fmaf(fabsf(x0.w - m3.x), i3.x, a3);
      a3 = fmaf(fabsf(x1.w - m3.y), i3.y, a3);
      a3 = fmaf(fabsf(x2.w - m3.z), i3.z, a3);
      a3 = fmaf(fabsf(x3.w - m3.w), i3.w, a3);
    }
    float w0 = validt[g0], w1 = validt[g1], w2 = validt[g2], w3 = validt[g3];
    tsum += w0 * a0 + w1 * a1 + w2 * a2 + w3 * a3;
    tcnt += (w0 + w1) + (w2 + w3);
  }

  // Deterministic block reduction: wave32 shuffle -> 8 partials -> thread 0.
  float s = wave_sum32(tsum);
  float c = wave_sum32(tcnt);
  int wid = tid >> 5, lane = tid & 31;
  if (lane == 0) { redS[wid] = s; redC[wid] = c; }
  __syncthreads();
  if (tid == 0) {
    float S = 0.0f, C = 0.0f;
    #pragma unroll
    for (int i = 0; i < NTHR / 32; ++i) { S += redS[i]; C += redC[i]; }
    partS[blockIdx.x] = S;
    partC[blockIdx.x] = C;
  }
}

// ---- Kernel 3: WMMA reduction of block partials (single wave) ---------------
// A = all-ones 16x4, so D[m,n] = sum_k B[k,n] + C[m,n]; summing row 0 of the
// accumulated D over all 16 columns yields the total regardless of the exact
// B-operand lane packing (any bijection of the 64 loaded values works).
// NBLK is a multiple of 64, so loads are unconditional (no exec juggling) and
// the loop fully unrolls: 8 tiles x 2 accumulating WMMAs, EXEC all-ones.
__global__ void mds_finalize(const float* __restrict__ partS,
                             const float* __restrict__ partC,
                             float* __restrict__ out) {
  int lane = threadIdx.x;           // 0..31, one full wave
  v8f cs = {}; v8f cc = {};
  v2f ones; ones.x = 1.0f; ones.y = 1.0f;
  #pragma unroll
  for (int i = 0; i < NBLK; i += 64) {
    v2f bs, bc;
    bs.x = partS[i + lane];
    bs.y = partS[i + 32 + lane];
    bc.x = partC[i + lane];
    bc.y = partC[i + 32 + lane];
    cs = __builtin_amdgcn_wmma_f32_16x16x4_f32(false, ones, false, bs,
                                               (short)0, cs, false, false);
    cc = __builtin_amdgcn_wmma_f32_16x16x4_f32(false, ones, false, bc,
                                               (short)0, cc, false, false);
  }
  // VGPR0 of D holds rows M=0 (lanes 0-15) and M=8 (lanes 16-31); rows are
  // identical (ones-A), so the full-wave sum counts each column twice.
  float s = wave_sum32(cs[0]) * 0.5f;
  float c = wave_sum32(cc[0]) * 0.5f;
  if (lane == 0) out[0] = s / (fmaxf(c, 1.0f) * (float)DD);
}

extern "C" void kernel_launch(void* const* d_in, const int* in_sizes, int n_in,
                              void* d_out, int out_size, void* d_ws, size_t ws_size,
                              hipStream_t stream) {
  const float* emb    = (const float*)d_in[0];   // [8,128,256,256]
  const float* logits = (const float*)d_in[1];   // [8,64,256,256]
  const float* pf     = (const float*)d_in[2];   // [64,128]
  const float* pc     = (const float*)d_in[3];   // [64]
  const float* var    = (const float*)d_in[4];   // [64,128,128]

  float* ws     = (float*)d_ws;
  float* inv_ev = ws;                    // 8192 floats
  float* partS  = ws + KK * DD;          // NBLK floats
  float* partC  = partS + NBLK;          // NBLK floats

  mds_inv_diag<<<(KK * DD) / 256, 256, 0, stream>>>(var, inv_ev);

  size_t shmem = (size_t)(2 * KK * STRIDE + KK + 16) * sizeof(float);
  mds_main<<<NBLK, NTHR, shmem, stream>>>(emb, logits, pf, pc, inv_ev,
                                          partS, partC);

  mds_finalize<<<1, 32, 0, stream>>>(partS, partC, (float*)d_out);
}